// HoughVotingLayer_86792699117605
// MI455X (gfx1250) — compile-verified
//
#include <hip/hip_runtime.h>

// Problem constants (match reference)
#define N_ 16
#define P_ 2048
#define K_ 512
#define H_ 256
#define W_ 256
#define HW_ (H_ * W_)
#define ZTHRESH_ 2.0f
#define MULT_ 5
#define PSPLIT_ 8                          // split P across 8 partial-vote waves per k-tile
#define PTILES_PER_ ((P_ / 16) / PSPLIT_)  // 16 p-tiles of 16 columns each

typedef __attribute__((ext_vector_type(2))) float v2f;
typedef __attribute__((ext_vector_type(8))) float v8f;

// ---------------------------------------------------------------------------
// Kernel 1: per (n,p) gather uv at pts and PRE-PACK the WMMA B-matrix in the
// exact per-lane layout of V_WMMA_F32_16X16X4_F32 (4x16 f32, K split across
// wave halves):
//   half 0 (lanes 0-15):  rows K=0,1  -> (u0, u1)
//   half 1 (lanes 16-31): rows K=2,3  -> (-c, 0),  c = pt . uv_pt
// Layout: Bpk[((n*2 + half)*P_ + p)] as float2.  Also zero the counters.
// ---------------------------------------------------------------------------
__global__ void __launch_bounds__(256)
prep_kernel(const float* __restrict__ uv, const int* __restrict__ pts,
            float* __restrict__ Bpk, int* __restrict__ wcnt, int* __restrict__ hcnt) {
    int tid = blockIdx.x * blockDim.x + threadIdx.x;
    if (tid < N_ * K_) { wcnt[tid] = 0; hcnt[tid] = 0; }
    if (tid >= N_ * P_) return;
    int n  = tid / P_;
    int p  = tid % P_;
    int py = pts[tid * 2 + 0];
    int px = pts[tid * 2 + 1];
    int flat = py * W_ + px;
    float u0 = uv[n * 2 * HW_ + flat];         // channel 0
    float u1 = uv[n * 2 * HW_ + HW_ + flat];   // channel 1
    float c  = (float)py * u0 + (float)px * u1;
    float* lo = Bpk + ((size_t)(n * 2 + 0) * P_ + p) * 2;
    float* hi = Bpk + ((size_t)(n * 2 + 1) * P_ + p) * 2;
    lo[0] = u0;  lo[1] = u1;
    hi[0] = -c;  hi[1] = 0.0f;
}

__device__ inline float block_reduce_512(float v, float* red, int k) {
    red[k] = v;
    __syncthreads();
#pragma unroll
    for (int s = 256; s > 0; s >>= 1) {
        if (k < s) red[k] += red[k + s];
        __syncthreads();
    }
    float r = red[0];
    __syncthreads();
    return r;
}

// ---------------------------------------------------------------------------
// Kernel 2: one block per batch n, one thread per k. 2x2 solve -> Y, then
// block-wide mean/std (ddof=1) -> z-score -> bad mask -> Yz.  NaNs propagate
// through the sums exactly as in the reference.
// ---------------------------------------------------------------------------
__global__ void __launch_bounds__(512)
solveY_kernel(const float* __restrict__ uv, const int* __restrict__ pts,
              const int* __restrict__ pair,
              float* __restrict__ Yv, int* __restrict__ iYv,
              float* __restrict__ Yz, int* __restrict__ bad) {
    __shared__ float red[512];
    int n = blockIdx.x, k = threadIdx.x;
    int idx = n * K_ + k;
    int i0 = pair[idx * 2 + 0];
    int i1 = pair[idx * 2 + 1];
    int p0c0 = pts[(n * P_ + i0) * 2 + 0], p0c1 = pts[(n * P_ + i0) * 2 + 1];
    int p1c0 = pts[(n * P_ + i1) * 2 + 0], p1c1 = pts[(n * P_ + i1) * 2 + 1];
    const float* uvn = uv + n * 2 * HW_;
    int f0 = p0c0 * W_ + p0c1;
    int f1 = p1c0 * W_ + p1c1;
    float uv0c0 = uvn[f0],       uv0c1 = uvn[HW_ + f0];
    float uv1c0 = uvn[f1],       uv1c1 = uvn[HW_ + f1];
    // A = [[uv0c0, -uv1c0], [uv0c1, -uv1c1]],  Bv = p1 - p0;  x0 from 2x2 inverse
    float a00 = uv0c0, a01 = -uv1c0, a10 = uv0c1, a11 = -uv1c1;
    float d0 = (float)(p1c0 - p0c0), d1 = (float)(p1c1 - p0c1);
    float det = a00 * a11 - a01 * a10;
    float x0  = (a11 * d0 - a01 * d1) / det;
    float Y0 = x0 * uv0c0 + (float)p0c0;
    float Y1 = x0 * uv0c1 + (float)p0c1;
    Yv[idx * 2 + 0] = Y0;
    Yv[idx * 2 + 1] = Y1;
    iYv[idx * 2 + 0] = (int)Y0;   // trunc toward zero, same as astype(int32)
    iYv[idx * 2 + 1] = (int)Y1;
    // mean / std (ddof=1), two-pass for fidelity
    float s0 = block_reduce_512(Y0, red, k);
    float s1 = block_reduce_512(Y1, red, k);
    float m0 = s0 / (float)K_, m1 = s1 / (float)K_;
    float e0 = Y0 - m0, e1 = Y1 - m1;
    float q0 = block_reduce_512(e0 * e0, red, k);
    float q1 = block_reduce_512(e1 * e1, red, k);
    float sd0 = sqrtf(q0 / (float)(K_ - 1));
    float sd1 = sqrtf(q1 / (float)(K_ - 1));
    float z0 = fabsf(e0 / sd0);
    float z1 = fabsf(e1 / sd1);
    int b = (z0 > ZTHRESH_) || (z1 > ZTHRESH_) || (Y0 != Y0) || (Y1 != Y1);
    bad[idx] = b;
    Yz[idx * 2 + 0] = b ? 0.0f : Y0;
    Yz[idx * 2 + 1] = b ? 0.0f : Y1;
}

// ---------------------------------------------------------------------------
// Kernel 3: the voting GEMM.  One wave per (n, k-tile, p-chunk).
// Score S[k][p] = Y0*u0 + Y1*u1 - c  via V_WMMA_F32_16X16X4_F32 with the
// pre-packed B matrix (no divergence in the inner loop).  Unrolled x4 so 4
// independent wmma+load chains are in flight per loop body.
// ---------------------------------------------------------------------------
__global__ void __launch_bounds__(256)
vote_kernel(const float* __restrict__ Yv, const int* __restrict__ iYv,
            const float* __restrict__ Bpk, const int* __restrict__ pts,
            int* __restrict__ wcnt, int* __restrict__ hcnt) {
    int lane = threadIdx.x & 31;
    int wid  = (blockIdx.x * blockDim.x + threadIdx.x) >> 5;
    int n     = wid / ((K_ / 16) * PSPLIT_);
    int rem   = wid % ((K_ / 16) * PSPLIT_);
    int ktile = rem / PSPLIT_;
    int pc    = rem % PSPLIT_;
    int k0    = ktile * 16;
    int hi    = lane >> 4;      // which half of the wave
    int lcol  = lane & 15;      // column within the 16-wide tile

    // A matrix: lanes 0-15 hold K=0,1 (= Y0,Y1 of row m=lane);
    //           lanes 16-31 hold K=2,3 (= constant 1, 0)
    v2f a;
    if (lane < 16) {
        a.x = Yv[(n * K_ + k0 + lane) * 2 + 0];
        a.y = Yv[(n * K_ + k0 + lane) * 2 + 1];
    } else {
        a.x = 1.0f;
        a.y = 0.0f;
    }

    // Integer Y for the 8 rows this lane's D elements belong to (m = hi*8 + j)
    int iy0[8], iy1[8];
#pragma unroll
    for (int j = 0; j < 8; j++) {
        int row = n * K_ + k0 + hi * 8 + j;
        iy0[j] = iYv[row * 2 + 0];
        iy1[j] = iYv[row * 2 + 1];
    }

    // Per-lane B base pointer: half 0 -> (u0,u1) rows, half 1 -> (-c,0) rows.
    const v2f* Bbase = (const v2f*)(Bpk + (size_t)(n * 2 + hi) * P_ * 2);
    const int2* ptsn = (const int2*)(pts + n * P_ * 2);

    int cnt[8] = {0, 0, 0, 0, 0, 0, 0, 0};
    int hc [8] = {0, 0, 0, 0, 0, 0, 0, 0};

    int pbase = pc * PTILES_PER_ * 16 + lcol;
#pragma unroll 4
    for (int t = 0; t < PTILES_PER_; t++) {
        int col = pbase + t * 16;
        v2f  b  = Bbase[col];      // unconditional global_load_b64
        int2 pt = ptsn[col];       // (py, px) for the integer-match count

        v8f acc = {};
        acc = __builtin_amdgcn_wmma_f32_16x16x4_f32(
            /*neg_a=*/false, a, /*neg_b=*/false, b,
            /*c_mod=*/(short)0, acc, /*reuse_a=*/false, /*reuse_b=*/false);

#pragma unroll
        for (int j = 0; j < 8; j++) {
            cnt[j] += (acc[j] > 0.0f) ? 1 : 0;                           // vote
            hc[j]  += ((iy0[j] == pt.x) && (iy1[j] == pt.y)) ? 1 : 0;    // int match
        }
    }

    // Reduce across the 16-lane group (xor masks 1,2,4,8 stay within each half)
#pragma unroll
    for (int j = 0; j < 8; j++) {
#pragma unroll
        for (int m = 1; m <= 8; m <<= 1) {
            cnt[j] += __shfl_xor(cnt[j], m, 32);
            hc[j]  += __shfl_xor(hc[j], m, 32);
        }
    }
    if (lcol == 0) {
#pragma unroll
        for (int j = 0; j < 8; j++) {
            atomicAdd(&wcnt[n * K_ + k0 + hi * 8 + j], cnt[j]);
            atomicAdd(&hcnt[n * K_ + k0 + hi * 8 + j], hc[j]);
        }
    }
}

// ---------------------------------------------------------------------------
// Kernel 4: weights, normalization, weighted mean, reversed output.
// ---------------------------------------------------------------------------
__global__ void __launch_bounds__(512)
finalize_kernel(const float* __restrict__ Yz, const int* __restrict__ bad,
                const int* __restrict__ wcnt, const int* __restrict__ hcnt,
                float* __restrict__ out) {
    __shared__ float red[512];
    int n = blockIdx.x, k = threadIdx.x;
    int idx = n * K_ + k;
    int factor = (hcnt[idx] == 1) ? MULT_ : 1;
    float w = bad[idx] ? 0.0f : (float)(factor * wcnt[idx]);
    float y0 = Yz[idx * 2 + 0];
    float y1 = Yz[idx * 2 + 1];
    float sw = block_reduce_512(w, red, k);
    float s0 = block_reduce_512(w * y0, red, k);
    float s1 = block_reduce_512(w * y1, red, k);
    if (k == 0) {
        float denom = fmaxf(sw, 1.0f);
        out[n * 2 + 0] = s1 / denom;   // pixel_xy = weighted_mean[:, ::-1]
        out[n * 2 + 1] = s0 / denom;
    }
}

// ---------------------------------------------------------------------------
extern "C" void kernel_launch(void* const* d_in, const int* in_sizes, int n_in,
                              void* d_out, int out_size, void* d_ws, size_t ws_size,
                              hipStream_t stream) {
    const float* uv   = (const float*)d_in[0];  // (N,2,H,W) f32
    const int*   pts  = (const int*)d_in[1];    // (N,P,2)   i32
    const int*   pair = (const int*)d_in[2];    // (N,K,2)   i32
    float* out = (float*)d_out;                 // (N,2)     f32

    char* ws = (char*)d_ws;
    size_t off = 0;
    auto carve = [&](size_t bytes) -> void* {
        void* p = ws + off;
        off += (bytes + 255) & ~(size_t)255;
        return p;
    };
    float* Yv   = (float*)carve(N_ * K_ * 2 * sizeof(float));
    int*   iYv  = (int*)  carve(N_ * K_ * 2 * sizeof(int));
    float* Yz   = (float*)carve(N_ * K_ * 2 * sizeof(float));
    int*   bad  = (int*)  carve(N_ * K_ * sizeof(int));
    float* Bpk  = (float*)carve((size_t)N_ * 2 * P_ * 2 * sizeof(float));
    int*   wcnt = (int*)  carve(N_ * K_ * sizeof(int));
    int*   hcnt = (int*)  carve(N_ * K_ * sizeof(int));
    (void)in_sizes; (void)n_in; (void)out_size; (void)ws_size;

    prep_kernel<<<(N_ * P_ + 255) / 256, 256, 0, stream>>>(uv, pts, Bpk, wcnt, hcnt);
    solveY_kernel<<<N_, 512, 0, stream>>>(uv, pts, pair, Yv, iYv, Yz, bad);
    int total_waves = N_ * (K_ / 16) * PSPLIT_;          // 4096 waves
    vote_kernel<<<total_waves / 8, 256, 0, stream>>>(Yv, iYv, Bpk, pts, wcnt, hcnt);
    finalize_kernel<<<N_, 512, 0, stream>>>(Yz, bad, wcnt, hcnt, out);
}